// GraphormerFishAttention_15779709846383
// MI455X (gfx1250) — compile-verified
//
#include <hip/hip_runtime.h>
#include <math.h>

// ---- problem constants ----
#define G_   8
#define L_   16
#define H_   768
#define D_   96      // H/G
#define B_   16
#define N_   512
#define QKVW 3072    // 768(q)+768(k)+1536(v)
#define LD_  1536    // L*D

typedef float v2f __attribute__((ext_vector_type(2)));
typedef float v8f __attribute__((ext_vector_type(8)));

// D(16x16,f32) = A(16x4,f32) * B(4x16,f32) + C   -- CDNA5 V_WMMA_F32_16X16X4_F32
static __device__ __forceinline__ v8f wmma4(v2f a, v2f b, v8f c) {
  return __builtin_amdgcn_wmma_f32_16x16x4_f32(
      /*neg_a=*/false, a, /*neg_b=*/false, b,
      /*c_mod=*/(short)0, c, /*reuse_a=*/false, /*reuse_b=*/false);
}

// Fragment conventions (ISA 7.12.2, 32-bit A 16x4 / C 16x16):
//  A: lane holds M = lane&15; VGPR j holds K = (lane>>4)*2 + j
//  B: lane holds N = lane&15; VGPR j holds K = (lane>>4)*2 + j
//  C: VGPR r holds M = r + 8*(lane>>4), N = lane&15

// ============================================================
// Kernel 1: QKV projection.  qkv[row, 0:3072] = x(8192x768) @ [Wq|Wk|Wv] (+bv)
// Wave computes 16 rows x 64 cols; 512*48 = 24576 waves.
// ============================================================
__global__ __launch_bounds__(128)
void k_proj(const float* __restrict__ x, const float* __restrict__ Wq,
            const float* __restrict__ Wk, const float* __restrict__ Wv,
            const float* __restrict__ bvp, float* __restrict__ qkv) {
  const int lane = threadIdx.x & 31, wvid = threadIdx.x >> 5;
  const int gw = blockIdx.x * 4 + wvid;
  const int mt = gw / 48;           // row tile (16 rows)
  const int ct = gw % 48;           // 64-col group
  const int col0 = ct * 64;

  const float* W; int ldw, wc0; bool isv = false;
  if (col0 < 768)        { W = Wq; ldw = 768;  wc0 = col0;        }
  else if (col0 < 1536)  { W = Wk; ldw = 768;  wc0 = col0 - 768;  }
  else                   { W = Wv; ldw = 1536; wc0 = col0 - 1536; isv = true; }

  const int row = mt * 16 + (lane & 15);
  const int klo = (lane >> 4) * 2;
  const float* xrow = x + (size_t)row * H_;

  v8f c0{}, c1{}, c2{}, c3{};
  for (int k0 = 0; k0 < H_; k0 += 4) {
    v2f a = *(const v2f*)(xrow + k0 + klo);
    const float* w0 = W + (size_t)(k0 + klo) * ldw + wc0 + (lane & 15);
    const float* w1 = w0 + ldw;
    v2f b;
    b.x = w0[0];  b.y = w1[0];  c0 = wmma4(a, b, c0);
    b.x = w0[16]; b.y = w1[16]; c1 = wmma4(a, b, c1);
    b.x = w0[32]; b.y = w1[32]; c2 = wmma4(a, b, c2);
    b.x = w0[48]; b.y = w1[48]; c3 = wmma4(a, b, c3);
  }

  float b0 = 0, b1 = 0, b2 = 0, b3 = 0;
  if (isv) {
    const float* bb = bvp + wc0 + (lane & 15);
    b0 = bb[0]; b1 = bb[16]; b2 = bb[32]; b3 = bb[48];
  }
  const int n0 = col0 + (lane & 15);
  const int r0 = mt * 16 + 8 * (lane >> 4);
  #pragma unroll
  for (int r = 0; r < 8; ++r) {
    float* orow = qkv + (size_t)(r0 + r) * QKVW;
    orow[n0     ] = c0[r] + b0;
    orow[n0 + 16] = c1[r] + b1;
    orow[n0 + 32] = c2[r] + b2;
    orow[n0 + 48] = c3[r] + b3;
  }
}

// ============================================================
// Kernel 2: gk[b,g,n,m] = sum_d Q[b,n,g,d]*K[b,m,g,d]   (512x512x96 per (b,g))
// Wave computes 16(n) x 64(m); 128 * 256 = 32768 waves.
// ============================================================
__global__ __launch_bounds__(128)
void k_gk(const float* __restrict__ qkv, float* __restrict__ gk) {
  const int lane = threadIdx.x & 31, wvid = threadIdx.x >> 5;
  const int gw = blockIdx.x * 4 + wvid;
  const int bg = gw >> 8;           // 256 waves per (b,g)
  const int t  = gw & 255;
  const int nt = t >> 3, mt = t & 7;
  const int b  = bg >> 3, g = bg & 7;

  const float* Q = qkv + (size_t)b * N_ * QKVW + g * D_;
  const float* K = Q + 768;
  const int n   = nt * 16 + (lane & 15);
  const int klo = (lane >> 4) * 2;
  const float* qrow = Q + (size_t)n * QKVW;

  v8f c[4] = {};
  for (int k0 = 0; k0 < D_; k0 += 4) {
    v2f a = *(const v2f*)(qrow + k0 + klo);
    #pragma unroll
    for (int j = 0; j < 4; ++j) {
      const int m = mt * 64 + j * 16 + (lane & 15);
      // B[k][m] = K[m][k] : contiguous along k -> 8B vector load
      v2f bf = *(const v2f*)(K + (size_t)m * QKVW + k0 + klo);
      c[j] = wmma4(a, bf, c[j]);
    }
  }
  float* out = gk + (((size_t)(b * G_ + g)) * N_ + nt * 16 + 8 * (lane >> 4)) * N_
                  + mt * 64 + (lane & 15);
  #pragma unroll
  for (int r = 0; r < 8; ++r)
    #pragma unroll
    for (int j = 0; j < 4; ++j)
      out[(size_t)r * N_ + j * 16] = c[j][r];
}

// ============================================================
// Block reduction helpers (512 threads = 16 waves, wave32)
// ============================================================
static __device__ __forceinline__ float blockMax(float v, volatile float* red,
                                                 int lane, int wvid) {
  #pragma unroll
  for (int o = 16; o; o >>= 1) v = fmaxf(v, __shfl_xor(v, o, 32));
  if (lane == 0) red[wvid] = v;
  __syncthreads();
  float r = red[0];
  #pragma unroll
  for (int i = 1; i < 16; ++i) r = fmaxf(r, red[i]);
  __syncthreads();
  return r;
}
static __device__ __forceinline__ float blockSum(float v, volatile float* red,
                                                 int lane, int wvid) {
  #pragma unroll
  for (int o = 16; o; o >>= 1) v += __shfl_xor(v, o, 32);
  if (lane == 0) red[wvid] = v;
  __syncthreads();
  float r = 0.f;
  #pragma unroll
  for (int i = 0; i < 16; ++i) r += red[i];
  __syncthreads();
  return r;
}
static __device__ __forceinline__ unsigned blockAnd(unsigned v, volatile unsigned* red,
                                                    int lane, int wvid) {
  #pragma unroll
  for (int o = 16; o; o >>= 1) v &= (unsigned)__shfl_xor((int)v, o, 32);
  if (lane == 0) red[wvid] = v;
  __syncthreads();
  unsigned r = 0xFFFFFFFFu;
  #pragma unroll
  for (int i = 0; i < 16; ++i) r &= red[i];
  __syncthreads();
  return r;
}

// ============================================================
// Kernel 3: fused mask -> +sigma^2*eps -> G->L mix -> mask2 -> mish
//           -> *scale + bias -> softmax(m) -> nan_to_num
// One 512-thread block per (b,n) row; thread owns one m, all 16 l-channels.
// ============================================================
__global__ __launch_bounds__(512)
void k_mix_softmax(const float* __restrict__ gk, const float* __restrict__ eps,
                   const float* __restrict__ ebias, const float* __restrict__ sigma,
                   const float* __restrict__ p, float* __restrict__ A) {
  __shared__ float    red[16];
  __shared__ unsigned redu[16];
  __shared__ float    sp[G_ * L_];
  __shared__ float    ssig2[G_];

  const int m = threadIdx.x;
  const int lane = m & 31, wvid = m >> 5;
  const int b = blockIdx.x >> 9;
  const int n = blockIdx.x & (N_ - 1);

  if (m < G_ * L_) sp[m] = p[m];
  if (m < G_)      ssig2[m] = sigma[m] * sigma[m];
  __syncthreads();

  // load gk row values, detect all-zero rows per g (pad_mask)
  float gkv[G_];
  unsigned zm = 0;
  #pragma unroll
  for (int g = 0; g < G_; ++g) {
    gkv[g] = gk[(((size_t)(b * G_ + g)) * N_ + n) * N_ + m];
    if (gkv[g] == 0.0f) zm |= (1u << g);
  }
  zm = blockAnd(zm, redu, lane, wvid);

  // a_g = pad ? 0 : gk + sigma^2 * eps
  float val[G_];
  #pragma unroll
  for (int g = 0; g < G_; ++g) {
    float e = eps[(((size_t)(b * G_ + g)) * N_ + n) * N_ + m];
    val[g] = ((zm >> g) & 1u) ? 0.0f : fmaf(ssig2[g], e, gkv[g]);
  }

  // mix groups into L channels: ap[l] = sum_g val[g]*p[g,l]
  float ap[L_];
  #pragma unroll
  for (int l = 0; l < L_; ++l) {
    float s = 0.f;
    #pragma unroll
    for (int g = 0; g < G_; ++g) s = fmaf(val[g], sp[g * L_ + l], s);
    ap[l] = s;
  }

  // pad_mask2: all-zero mixed rows
  unsigned z2 = 0;
  #pragma unroll
  for (int l = 0; l < L_; ++l)
    if (ap[l] == 0.0f) z2 |= (1u << l);
  z2 = blockAnd(z2, redu, lane, wvid);

  const float bv   = ebias[((size_t)b * N_ + n) * N_ + m];
  const float scale = 0.03608439182435161f;  // 768^-0.5

  #pragma unroll 1
  for (int l = 0; l < L_; ++l) {
    float t;
    if ((z2 >> l) & 1u) {
      t = -INFINITY;
    } else {
      float xx = ap[l];
      float spl = (xx > 20.f) ? xx : log1pf(__expf(xx));  // softplus
      t = xx * tanhf(spl);                                 // mish
    }
    t = fmaf(t, scale, bv);
    float mx = blockMax(t, red, lane, wvid);
    float e  = __expf(t - mx);          // all -inf row -> exp(NaN) -> NaN
    float sm = blockSum(e, red, lane, wvid);
    float pv = e / sm;
    if (pv != pv) pv = 0.0f;            // nan_to_num
    A[(((size_t)(b * L_ + l)) * N_ + n) * N_ + m] = pv;
  }
}

// ============================================================
// Kernel 4: o[b,n,l*96+d] = sum_m A[b,l,n,m] * V[b,m,l,d]  (512x96x512 per (b,l))
// Wave computes 16(n) x 96(d); 256 * 32 = 8192 waves.
// ============================================================
__global__ __launch_bounds__(128)
void k_av(const float* __restrict__ A, const float* __restrict__ qkv,
          float* __restrict__ o) {
  const int lane = threadIdx.x & 31, wvid = threadIdx.x >> 5;
  const int gw = blockIdx.x * 4 + wvid;
  const int bl = gw >> 5, nt = gw & 31;
  const int b = bl >> 4, l = bl & 15;

  const float* Ab = A + ((size_t)(b * L_ + l)) * N_ * N_;
  const float* V  = qkv + (size_t)b * N_ * QKVW + LD_ + l * D_;
  const int n   = nt * 16 + (lane & 15);
  const int klo = (lane >> 4) * 2;
  const float* arow = Ab + (size_t)n * N_;

  v8f c[6] = {};
  for (int k0 = 0; k0 < N_; k0 += 4) {
    v2f a = *(const v2f*)(arow + k0 + klo);
    #pragma unroll
    for (int j = 0; j < 6; ++j) {
      const int d = j * 16 + (lane & 15);
      const float* vp = V + (size_t)(k0 + klo) * QKVW + d;
      v2f bf; bf.x = vp[0]; bf.y = vp[QKVW];
      c[j] = wmma4(a, bf, c[j]);
    }
  }
  float* orow = o + ((size_t)b * N_ + nt * 16 + 8 * (lane >> 4)) * LD_
                  + l * D_ + (lane & 15);
  #pragma unroll
  for (int r = 0; r < 8; ++r)
    #pragma unroll
    for (int j = 0; j < 6; ++j)
      orow[(size_t)r * LD_ + j * 16] = c[j][r];
}

// ============================================================
// Kernel 5: out(8192x768) = o(8192x1536) @ Wout(1536x768)
// Wave computes 16 rows x 64 cols; 512*12 = 6144 waves.
// ============================================================
__global__ __launch_bounds__(128)
void k_out(const float* __restrict__ o, const float* __restrict__ Wout,
           float* __restrict__ out) {
  const int lane = threadIdx.x & 31, wvid = threadIdx.x >> 5;
  const int gw = blockIdx.x * 4 + wvid;
  const int mt = gw / 12, ct = gw % 12;
  const int row = mt * 16 + (lane & 15);
  const int klo = (lane >> 4) * 2;
  const float* orow = o + (size_t)row * LD_;

  v8f c[4] = {};
  for (int k0 = 0; k0 < LD_; k0 += 4) {
    v2f a = *(const v2f*)(orow + k0 + klo);
    const float* w0 = Wout + (size_t)(k0 + klo) * H_ + ct * 64 + (lane & 15);
    #pragma unroll
    for (int j = 0; j < 4; ++j) {
      v2f bf; bf.x = w0[j * 16]; bf.y = w0[j * 16 + H_];
      c[j] = wmma4(a, bf, c[j]);
    }
  }
  float* dst = out + (size_t)(mt * 16 + 8 * (lane >> 4)) * H_ + ct * 64 + (lane & 15);
  #pragma unroll
  for (int r = 0; r < 8; ++r)
    #pragma unroll
    for (int j = 0; j < 4; ++j)
      dst[(size_t)r * H_ + j * 16] = c[j][r];
}

// ============================================================
extern "C" void kernel_launch(void* const* d_in, const int* in_sizes, int n_in,
                              void* d_out, int out_size, void* d_ws, size_t ws_size,
                              hipStream_t stream) {
  const float* x    = (const float*)d_in[0];
  const float* eb   = (const float*)d_in[1];
  const float* eps  = (const float*)d_in[2];
  const float* Wq   = (const float*)d_in[3];
  const float* Wk   = (const float*)d_in[4];
  const float* Wv   = (const float*)d_in[5];
  const float* bv   = (const float*)d_in[6];
  const float* sg   = (const float*)d_in[7];
  const float* p    = (const float*)d_in[8];
  const float* Wout = (const float*)d_in[9];
  float* out = (float*)d_out;

  float* ws   = (float*)d_ws;
  float* qkv  = ws;                                   // B*N*3072   = 25.2M floats
  float* gk   = qkv  + (size_t)B_ * N_ * QKVW;        // B*G*N*N    = 33.6M floats
  float* Abuf = gk   + (size_t)B_ * G_ * N_ * N_;     // B*L*N*N    = 67.1M floats
  float* obuf = gk;  // reuse gk region for o (12.6M floats) after k_mix_softmax

  k_proj<<<6144, 128, 0, stream>>>(x, Wq, Wk, Wv, bv, qkv);
  k_gk<<<8192, 128, 0, stream>>>(qkv, gk);
  k_mix_softmax<<<B_ * N_, 512, 0, stream>>>(gk, eps, eb, sg, p, Abuf);
  k_av<<<2048, 128, 0, stream>>>(Abuf, qkv, obuf);
  k_out<<<1536, 128, 0, stream>>>(obuf, Wout, out);
}